// CBDA_11562051961402
// MI455X (gfx1250) — compile-verified
//
#include <hip/hip_runtime.h>

typedef __attribute__((ext_vector_type(2))) float v2f;
typedef __attribute__((ext_vector_type(8))) float v8f;

#define B_    1024
#define T_    256
#define DCOND 768
#define DOUT  86
#define NCOLS 172   // 2*DOUT

// ---------------------------------------------------------------------------
// Kernel 1: gb[1024,172] = x_cond[1024,768] @ W[768,172] + bias
// One wave32 per 16x16 output tile, K marched in steps of 4 via
// V_WMMA_F32_16X16X4_F32 (f32 A/B/C — matches reference precision exactly).
// ---------------------------------------------------------------------------
__global__ __launch_bounds__(32)
void film_gemm_wmma(const float* __restrict__ A,     // x_cond (1024,768)
                    const float* __restrict__ W,     // (768,172)
                    const float* __restrict__ bias,  // (172,)
                    float* __restrict__ gb)          // (1024,172)
{
    const int lane = threadIdx.x;      // 0..31
    const int nt   = blockIdx.x;       // 0..10  (11 N-tiles, 176 >= 172)
    const int mt   = blockIdx.y;       // 0..63  (64 M-tiles, exact)
    const int l15  = lane & 15;
    const int half = lane >> 4;        // 0: K {0,1}, 1: K {2,3}

    const int m = mt * 16 + l15;                 // A row for this lane
    const int n = nt * 16 + l15;                 // B/D column for this lane
    const bool nvalid = (n < NCOLS);
    const int nc = nvalid ? n : (NCOLS - 1);     // clamp to stay in-bounds

    const float* __restrict__ Arow = A + m * DCOND;

    v8f acc = {};
    #pragma unroll 4
    for (int k0 = 0; k0 < DCOND; k0 += 4) {
        const int ka = k0 + 2 * half;            // even -> 8B-aligned float2
        v2f a;
        a.x = Arow[ka];
        a.y = Arow[ka + 1];

        float b0 = W[(ka    ) * NCOLS + nc];
        float b1 = W[(ka + 1) * NCOLS + nc];
        v2f bb;
        bb.x = nvalid ? b0 : 0.0f;
        bb.y = nvalid ? b1 : 0.0f;

        // 8 args: (neg_a, A, neg_b, B, c_mod, C, reuse_a, reuse_b)
        acc = __builtin_amdgcn_wmma_f32_16x16x4_f32(
            /*neg_a=*/false, a, /*neg_b=*/false, bb,
            /*c_mod=*/(short)0, acc, /*reuse_a=*/false, /*reuse_b=*/false);
    }

    // D layout (32-bit 16x16): VGPR r -> M = r + 8*half, N = lane&15
    if (nvalid) {
        const float bv   = bias[n];
        const int mbase  = mt * 16 + half * 8;
        #pragma unroll
        for (int r = 0; r < 8; ++r) {
            gb[(mbase + r) * NCOLS + n] = acc[r] + bv;
        }
    }
}

// ---------------------------------------------------------------------------
// Kernel 2: FiLM + block-diagonal scatter, fully store-bandwidth bound.
// out[b,t,c] = (c in [i*86, i*86+86) with i = t-range) ?
//              (1+gamma[b,j])*x[b,t,j] + beta[b,j]   (j = c - i*86)  : 0
// Block = (64,4): one batch b, 4 t-rows; gamma/beta staged in LDS;
// one float4 (16B-aligned) store per thread.
// ---------------------------------------------------------------------------
__global__ __launch_bounds__(256)
void film_apply(const float* __restrict__ xf,   // (1024,256,86)
                const float* __restrict__ gb,   // (1024,172)
                float* __restrict__ out)        // (1024,256,256)
{
    __shared__ float s_gamma[DOUT];
    __shared__ float s_beta[DOUT];

    const int b   = blockIdx.y;
    const int tid = threadIdx.y * 64 + threadIdx.x;
    if (tid < NCOLS) {
        const float v = gb[b * NCOLS + tid];
        if (tid < DOUT) s_gamma[tid]        = 1.0f + v;
        else            s_beta[tid - DOUT]  = v;
    }
    __syncthreads();

    const int t    = blockIdx.x * 4 + threadIdx.y;
    const int i    = (t < DOUT) ? 0 : ((t < 2 * DOUT) ? 1 : 2);
    const int base = i * DOUT;
    const int c0   = threadIdx.x * 4;

    const float* __restrict__ xrow = xf + (size_t)(b * T_ + t) * DOUT;

    float4 v = make_float4(0.0f, 0.0f, 0.0f, 0.0f);
    #pragma unroll
    for (int e = 0; e < 4; ++e) {
        const int j = c0 + e - base;
        if (j >= 0 && j < DOUT) {          // c<256 makes j<84 for i=2 automatic
            const float x = xrow[j];
            (&v.x)[e] = fmaf(s_gamma[j], x, s_beta[j]);
        }
    }
    *(float4*)(out + (((size_t)(b * T_ + t)) << 8) + c0) = v;
}

// ---------------------------------------------------------------------------
extern "C" void kernel_launch(void* const* d_in, const int* in_sizes, int n_in,
                              void* d_out, int out_size, void* d_ws, size_t ws_size,
                              hipStream_t stream) {
    (void)in_sizes; (void)n_in; (void)out_size; (void)ws_size;

    const float* x_cond    = (const float*)d_in[0];  // (1024,768)
    const float* x_to_film = (const float*)d_in[1];  // (1024,256,86)
    const float* W         = (const float*)d_in[2];  // (768,172)
    const float* bias      = (const float*)d_in[3];  // (172,)
    float*       out       = (float*)d_out;          // (1024,256,256)
    float*       gb        = (float*)d_ws;           // 1024*172 floats = 704 KB

    // GEMM: 11 N-tiles x 64 M-tiles, one wave each
    dim3 g1(11, 64);
    film_gemm_wmma<<<g1, dim3(32, 1, 1), 0, stream>>>(x_cond, W, bias, gb);

    // Epilogue: one block per (b, 4 t-rows)
    dim3 g2(T_ / 4, B_);
    dim3 blk(64, 4);
    film_apply<<<g2, blk, 0, stream>>>(x_to_film, gb, out);
}